// FPSsubsample_9723805958816
// MI455X (gfx1250) — compile-verified
//
#include <hip/hip_runtime.h>
#include <stdint.h>

#define FPS_BLOCK 1024
#define MAX_PPT 8            // supports n up to 8192
#define GATHER_BLOCK 512

// Monotone float -> uint key: unsigned compare == float compare.
__device__ __forceinline__ unsigned monokey(float f) {
  unsigned u = __float_as_uint(f);
  return (u & 0x80000000u) ? ~u : (u | 0x80000000u);
}

__device__ __forceinline__ unsigned long long shfl_xor_u64(unsigned long long v, int m) {
  unsigned lo = (unsigned)(v & 0xffffffffull);
  unsigned hi = (unsigned)(v >> 32);
  lo = __shfl_xor(lo, m, 32);   // wave32 on gfx1250
  hi = __shfl_xor(hi, m, 32);
  return ((unsigned long long)hi << 32) | lo;
}

// One workgroup per batch. Sequential FPS; distances (squared norms) in registers.
// Single barrier per step via rotating 3-slot LDS atomic-max accumulator;
// speculative per-wave candidate-row prefetch overlaps HBM latency with the barrier.
__global__ __launch_bounds__(FPS_BLOCK)
void fps_kernel(const float* __restrict__ ab, const unsigned char* __restrict__ mask,
                const int* __restrict__ start, int* __restrict__ qidx,
                int n, int m) {
  const int b = blockIdx.x;
  const int t = threadIdx.x;
  const int P = (n + FPS_BLOCK - 1) / FPS_BLOCK;   // points per thread (contiguous)
  const int j0 = t * P;

  __shared__ int s_scan[FPS_BLOCK];
  __shared__ unsigned long long s_slot[3];         // rotating argmax accumulators

  float dist[MAX_PPT];
  unsigned char msk[MAX_PPT];
  int cnt = 0;
  for (int p = 0; p < P; ++p) {
    int j = j0 + p;
    unsigned char mm = (j < n) ? mask[(size_t)b * n + j] : (unsigned char)0;
    msk[p] = mm;
    dist[p] = 1e16f;              // (1e8)^2 : squared-distance domain
    cnt += mm ? 1 : 0;
  }

  // ---- seed: (start % count)-th valid point. Hillis-Steele inclusive scan. ----
  s_scan[t] = cnt;
  if (t == 0) { s_slot[1] = 0ull; s_slot[2] = 0ull; }
  __syncthreads();
  for (int off = 1; off < FPS_BLOCK; off <<= 1) {
    int v = s_scan[t];
    int add = (t >= off) ? s_scan[t - off] : 0;
    __syncthreads();
    s_scan[t] = v + add;
    __syncthreads();
  }
  int total = s_scan[FPS_BLOCK - 1];
  int excl = s_scan[t] - cnt;
  if (total == 0) {
    if (t == 0) s_slot[0] = (unsigned long long)(unsigned)(n - 1);  // far = 0
  } else {
    int iw = start[b] % total;
    if (iw < 0) iw += total;
    if (iw >= excl && iw < excl + cnt) {
      int want = iw - excl, seen = 0;
      for (int p = 0; p < P; ++p) {
        if (msk[p]) {
          if (seen == want) {
            s_slot[0] = (unsigned long long)(unsigned)(n - 1 - (j0 + p));
            break;
          }
          ++seen;
        }
      }
    }
  }
  __syncthreads();

  // ---- main sequential FPS loop ----
  const size_t rowstride = (size_t)n * 3;
  const float* abb = ab + (size_t)b * n * rowstride;
  const int lane = t & 31;
  const unsigned rowbytes = (unsigned)(n * 3) * 4u;

  for (int s = 0; s < m; ++s) {
    const unsigned long long rk = s_slot[s % 3];            // prev-step result
    const int far = n - 1 - (int)(rk & 0xffffffffu);
    if (t == 0) qidx[(size_t)b * m + s] = far;              // chosen BEFORE update

    const float* row = abb + (size_t)far * rowstride + (size_t)3 * j0;
    unsigned long long best = 0ull;
    for (int p = 0; p < P; ++p) {
      int j = j0 + p;
      if (j < n) {
        float d;
        if (msk[p]) {
          float x = row[3 * p + 0], y = row[3 * p + 1], z = row[3 * p + 2];
          d = fminf(dist[p], x * x + y * y + z * z);  // squared norm: sqrt elided (monotone)
        } else {
          d = -100.0f;                                // min(prev, -100) == -100 always
        }
        dist[p] = d;
        unsigned long long key =
            ((unsigned long long)monokey(d) << 32) | (unsigned)(n - 1 - j); // tie -> lowest j
        best = (key > best) ? key : best;
      }
    }
    // wave32 butterfly: all lanes end with the wave-wide max
    for (int o = 16; o; o >>= 1) {
      unsigned long long ok = shfl_xor_u64(best, o);
      best = (ok > best) ? ok : best;
    }
    // speculative prefetch of this wave's candidate row: the global winner is one
    // of the 32 wave candidates, so its row is being pulled toward L2 during the
    // barrier + ds-atomic window below (one lane per 128B cacheline).
    {
      int cand = n - 1 - (int)(best & 0xffffffffu);
      const char* cbase = (const char*)(abb + (size_t)cand * rowstride);
      for (unsigned off = (unsigned)lane * 128u; off < rowbytes; off += 32u * 128u)
        __builtin_prefetch(cbase + off, 0, 1);
    }
    if (lane == 0)
      atomicMax(&s_slot[(s + 1) % 3], best);                // ds_max_u64
    if (t == 0)
      s_slot[(s + 2) % 3] = 0ull;                           // recycle (safe: disjoint roles)
    __syncthreads();                                        // single barrier per step
  }
}

// One workgroup per output row (b,i). Stage the contiguous 24KB source row
// ab_pairs[b, q_i, :, :] into LDS with gfx1250 async-to-LDS DMA, then gather
// the m selected columns from LDS (turns scattered 12B global gathers into
// full-bandwidth contiguous reads + LDS gathers) and write coalesced.
__global__ __launch_bounds__(GATHER_BLOCK)
void gather_kernel(const float* __restrict__ ab, const float* __restrict__ vals,
                   const unsigned char* __restrict__ mask, const int* __restrict__ qidx,
                   float* __restrict__ out_ab, float* __restrict__ out_vals,
                   float* __restrict__ out_mask, int n, int m, int c) {
  extern __shared__ float smem[];                 // [n*3] floats row, then [m] ints q
  float* s_row = smem;
  int* s_q = (int*)(smem + (size_t)n * 3);

  const int bi = blockIdx.x;
  const int b = bi / m, i = bi % m;
  const int tid = threadIdx.x;
  const int qi = qidx[(size_t)b * m + i];

  const float* src = ab + ((size_t)b * n + (size_t)qi) * (size_t)n * 3;
  const unsigned nbytes = (unsigned)(n * 3) * 4u;
  const unsigned nb16 = nbytes & ~15u;

#if defined(__gfx1250__)
  {
    unsigned lds_base = (unsigned)(uintptr_t)s_row;          // LDS byte address (low 32 bits)
    unsigned long long gbase = (unsigned long long)(uintptr_t)src;
    for (unsigned off = (unsigned)tid * 16u; off < nb16; off += (unsigned)GATHER_BLOCK * 16u) {
      unsigned laddr = lds_base + off;
      unsigned long long gaddr = gbase + (unsigned long long)off;
      asm volatile("global_load_async_to_lds_b128 %0, %1, off"
                   :: "v"(laddr), "v"(gaddr) : "memory");
    }
    for (unsigned off = nb16 + (unsigned)tid * 4u; off < nbytes; off += (unsigned)GATHER_BLOCK * 4u) {
      unsigned laddr = lds_base + off;
      unsigned long long gaddr = gbase + (unsigned long long)off;
      asm volatile("global_load_async_to_lds_b32 %0, %1, off"
                   :: "v"(laddr), "v"(gaddr) : "memory");
    }
  }
#else
  for (unsigned e = (unsigned)tid; e < (unsigned)(n * 3); e += (unsigned)GATHER_BLOCK)
    s_row[e] = src[e];
#endif

  for (int j = tid; j < m; j += GATHER_BLOCK)
    s_q[j] = qidx[(size_t)b * m + j];

#if defined(__gfx1250__)
  asm volatile("s_wait_asynccnt 0" ::: "memory");
#endif
  __syncthreads();

  float* oab = out_ab + (size_t)bi * (size_t)m * 3;
  for (int j = tid; j < m; j += GATHER_BLOCK) {
    int qj = s_q[j];
    float x = s_row[3 * qj + 0];
    float y = s_row[3 * qj + 1];
    float z = s_row[3 * qj + 2];
    oab[3 * j + 0] = x;
    oab[3 * j + 1] = y;
    oab[3 * j + 2] = z;
  }

  const float* vsrc = vals + ((size_t)b * n + (size_t)qi) * (size_t)c;
  float* ov = out_vals + (size_t)bi * (size_t)c;
  for (int j = tid; j < c; j += GATHER_BLOCK) ov[j] = vsrc[j];

  if (tid == 0) out_mask[bi] = mask[(size_t)b * n + qi] ? 1.0f : 0.0f;
}

extern "C" void kernel_launch(void* const* d_in, const int* in_sizes, int n_in,
                              void* d_out, int out_size, void* d_ws, size_t ws_size,
                              hipStream_t stream) {
  const float* ab = (const float*)d_in[0];              // (bs,n,n,3) f32
  const float* vals = (const float*)d_in[1];            // (bs,n,c)   f32
  const unsigned char* mask = (const unsigned char*)d_in[2]; // (bs,n) bool
  const int* start = (const int*)d_in[3];               // (bs,) int

  const int bs = in_sizes[3];
  const int n = in_sizes[2] / bs;
  const int c = in_sizes[1] / in_sizes[2];
  const int m = (int)((double)n * 0.25 + 0.5);

  int* qidx = (int*)d_ws;                               // bs*m ints of scratch

  float* out_ab = (float*)d_out;                        // (bs,m,m,3)
  float* out_vals = out_ab + (size_t)bs * m * m * 3;    // (bs,m,c)
  float* out_mask = out_vals + (size_t)bs * m * c;      // (bs,m)

  fps_kernel<<<bs, FPS_BLOCK, 0, stream>>>(ab, mask, start, qidx, n, m);

  size_t shmem = ((size_t)n * 3 + (size_t)m) * sizeof(float);
  gather_kernel<<<bs * m, GATHER_BLOCK, shmem, stream>>>(
      ab, vals, mask, qidx, out_ab, out_vals, out_mask, n, m, c);
}